// SBGNNLayer_61795989454873
// MI455X (gfx1250) — compile-verified
//
#include <hip/hip_runtime.h>
#include <hip/hip_bf16.h>

// ---------------------------------------------------------------------------
// SBGNN layer for MI455X (gfx1250, wave32).
// Dense transforms + update MLP run on V_WMMA_F32_16X16X4_F32 (native fp32
// matrix pipe). Edge aggregation: one wave32 per edge, coalesced gathers,
// wave-shuffle reduction, HW f32 global atomics for segment sums.
// ---------------------------------------------------------------------------

typedef __attribute__((ext_vector_type(2))) float v2f;
typedef __attribute__((ext_vector_type(8))) float v8f;

__device__ __forceinline__ v8f wmma_f32_k4(v2f a, v2f b, v8f c) {
    // 8 args: (neg_a, A, neg_b, B, c_mod, C, reuse_a, reuse_b)
    return __builtin_amdgcn_wmma_f32_16x16x4_f32(false, a, false, b,
                                                 (short)0, c, false, false);
}

// Y[n][j] = sum_k X[n][k] * W[j][k] + bias[j] ; X:[N,64] (reads clamped),
// Y padded to Ntiles*16 rows. One wave per 16x16 output tile.
__global__ void transform_kernel(const float* __restrict__ X,
                                 const float* __restrict__ W,
                                 const float* __restrict__ bias,
                                 float* __restrict__ Y,
                                 int N, int Ntiles) {
    int wv   = threadIdx.x >> 5;
    int lane = threadIdx.x & 31;
    int tile = blockIdx.x * 8 + wv;          // wave-uniform
    int total = Ntiles * 4;
    if (tile >= total) return;               // whole wave exits together
    int rt = tile >> 2, ct = tile & 3;
    int row0 = rt * 16, j0 = ct * 16;
    int half = lane >> 4;                    // 0: K even pair, 1: K odd pair
    int l    = lane & 15;
    int arow = row0 + l; if (arow >= N) arow = N - 1;

    v8f acc = {};
    #pragma unroll
    for (int k0 = 0; k0 < 64; k0 += 4) {
        int ka = k0 + half * 2;
        v2f a; a.x = X[arow * 64 + ka];          a.y = X[arow * 64 + ka + 1];
        v2f b; b.x = W[(j0 + l) * 64 + ka];      b.y = W[(j0 + l) * 64 + ka + 1];
        acc = wmma_f32_k4(a, b, acc);
    }
    float bj = bias[j0 + l];
    #pragma unroll
    for (int i = 0; i < 8; ++i) {
        int m = i + 8 * half;                 // C/D layout: VGPR i -> row i / i+8
        Y[(size_t)(row0 + m) * 64 + j0 + l] = acc[i] + bj;
    }
}

__global__ void zero_kernel(float* __restrict__ p, long long n) {
    long long i = (long long)blockIdx.x * blockDim.x + threadIdx.x;
    if (i < n) p[i] = 0.0f;
}

// One wave32 per edge. Lane d owns dims d and d+32.
// wmode: 0 = no weight, 1 = matrix[src*NBc+dst], 2 = matrix[dst*NBc+src]
__global__ void edge_agg_kernel(const int* __restrict__ edges, int E,
                                const float* __restrict__ feat_src,
                                const float* __restrict__ new_emb,
                                const float* __restrict__ a_vec,   // [128]
                                const float* __restrict__ matrix,
                                int wmode, int NBc,
                                float* __restrict__ num,
                                float* __restrict__ den) {
    int lane = threadIdx.x & 31;
    int e    = (int)(((long long)blockIdx.x * blockDim.x + threadIdx.x) >> 5);
    if (e >= E) return;
    int src = edges[2 * e], dst = edges[2 * e + 1];

    float w = 1.0f;
    if (wmode == 1)      w = matrix[(size_t)src * NBc + dst];
    else if (wmode == 2) w = matrix[(size_t)dst * NBc + src];

    int d0 = lane, d1 = lane + 32;
    float fb0 = new_emb[(size_t)dst * 64 + d0];
    float fb1 = new_emb[(size_t)dst * 64 + d1];
    float fa0 = feat_src[(size_t)src * 64 + d0];
    float fa1 = feat_src[(size_t)src * 64 + d1];

    float part = fa0 * a_vec[d0] + fa1 * a_vec[d1]
               + w * (fb0 * a_vec[64 + d0] + fb1 * a_vec[64 + d1]);
    #pragma unroll
    for (int off = 16; off > 0; off >>= 1)
        part += __shfl_xor(part, off, 32);

    float s  = part;
    float el = s > 0.0f ? s : 0.1f * expm1f(s);   // elu01
    float score = expf(el);

    unsafeAtomicAdd(&num[(size_t)src * 64 + d0], score * fb0);
    unsafeAtomicAdd(&num[(size_t)src * 64 + d1], score * fb1);
    if (lane == 0) unsafeAtomicAdd(&den[src], score);
}

__global__ void finalize_kernel(float* __restrict__ m,
                                const float* __restrict__ den, int Nseg) {
    long long i = (long long)blockIdx.x * blockDim.x + threadIdx.x;
    if (i >= (long long)Nseg * 64) return;
    float d = den[i >> 6];
    if (d == 0.0f) d = 1.0f;
    m[i] = m[i] / d;
}

// Fused update: 16 nodes per block (256 threads = 8 waves).
// x = [feat | m0 | m1 | m2 | m3] (320 wide) staged in LDS,
// h = prelu(x @ W1.T + b1) in LDS, out = h @ W2.T + b2.
__global__ void update_kernel(const float* __restrict__ feat,
                              const float* __restrict__ m0,
                              const float* __restrict__ m1,
                              const float* __restrict__ m2,
                              const float* __restrict__ m3,
                              const float* __restrict__ W1,
                              const float* __restrict__ b1,
                              const float* __restrict__ alpha_p,
                              const float* __restrict__ W2,
                              const float* __restrict__ b2,
                              float* __restrict__ out, int N) {
    __shared__ float x_s[16 * 321];   // +1 pad: stride 320 would be bank-aligned
    __shared__ float h_s[16 * 129];   // +1 pad: stride 128 would be bank-aligned

    int row0 = blockIdx.x * 16;
    const float* srcs[5] = {feat, m0, m1, m2, m3};

    for (int i = threadIdx.x; i < 16 * 320; i += 256) {
        int r = i / 320, c = i % 320;
        int buf = c >> 6, cc = c & 63;
        int rr = row0 + r;
        float v;
        if (buf == 0) {
            int cr = rr < N ? rr : N - 1;
            v = feat[(size_t)cr * 64 + cc];
        } else {
            v = srcs[buf][(size_t)rr * 64 + cc];   // m buffers are tile-padded
        }
        x_s[r * 321 + c] = v;
    }
    __syncthreads();

    int wv   = threadIdx.x >> 5;
    int lane = threadIdx.x & 31;
    int half = lane >> 4;
    int l    = lane & 15;
    float alpha = alpha_p[0];

    {   // h[:, wv*16 : wv*16+16]
        int j0 = wv * 16;
        v8f acc = {};
        #pragma unroll 8
        for (int k0 = 0; k0 < 320; k0 += 4) {
            int ka = k0 + half * 2;
            v2f a; a.x = x_s[l * 321 + ka];           a.y = x_s[l * 321 + ka + 1];
            v2f b; b.x = W1[(j0 + l) * 320 + ka];     b.y = W1[(j0 + l) * 320 + ka + 1];
            acc = wmma_f32_k4(a, b, acc);
        }
        float bj = b1[j0 + l];
        #pragma unroll
        for (int i = 0; i < 8; ++i) {
            int m = i + 8 * half;
            float h = acc[i] + bj;
            h = h >= 0.0f ? h : alpha * h;            // prelu
            h_s[m * 129 + j0 + l] = h;
        }
    }
    __syncthreads();

    if (wv < 4) {   // out[:, wv*16 : wv*16+16]
        int o0 = wv * 16;
        v8f acc = {};
        #pragma unroll 8
        for (int k0 = 0; k0 < 128; k0 += 4) {
            int ka = k0 + half * 2;
            v2f a; a.x = h_s[l * 129 + ka];           a.y = h_s[l * 129 + ka + 1];
            v2f b; b.x = W2[(o0 + l) * 128 + ka];     b.y = W2[(o0 + l) * 128 + ka + 1];
            acc = wmma_f32_k4(a, b, acc);
        }
        float bj = b2[o0 + l];
        #pragma unroll
        for (int i = 0; i < 8; ++i) {
            int m = i + 8 * half;
            int r = row0 + m;
            if (r < N) out[(size_t)r * 64 + o0 + l] = acc[i] + bj;
        }
    }
}

extern "C" void kernel_launch(void* const* d_in, const int* in_sizes, int n_in,
                              void* d_out, int out_size, void* d_ws, size_t ws_size,
                              hipStream_t stream) {
    const float* feature_a = (const float*)d_in[0];
    const float* feature_b = (const float*)d_in[1];
    const float* matrix    = (const float*)d_in[2];
    const float* W_agg     = (const float*)d_in[3];   // [8,64,64]
    const float* b_agg     = (const float*)d_in[4];   // [8,64]
    const float* a_agg     = (const float*)d_in[5];   // [8,128]
    const float* W1        = (const float*)d_in[6];   // [128,320]
    const float* b1        = (const float*)d_in[7];   // [128]
    const float* alpha     = (const float*)d_in[8];   // [1]
    const float* W2        = (const float*)d_in[9];   // [64,128]
    const float* b2        = (const float*)d_in[10];  // [64]

    const int NA = in_sizes[0] / 64;
    const int NB = in_sizes[1] / 64;
    const int tilesA = (NA + 15) / 16;
    const int tilesB = (NB + 15) / 16;
    const int PAD    = tilesB * 16;                   // max padded rows

    float* ws = (float*)d_ws;
    const size_t mstride = (size_t)PAD * 64;
    float* mbuf[8];
    for (int k = 0; k < 8; ++k) mbuf[k] = ws + (size_t)k * mstride;
    float* tmp_emb = ws + 8 * mstride;                // padded new_emb, reused
    float* den     = ws + 9 * mstride;                // [PAD]

    struct Cfg { int ei; const float* fsrc; const float* femb;
                 int Nseg; int Nemb; int wmode; };
    const Cfg cfg[8] = {
        {11, feature_a, feature_b, NA, NB, 1},  // m1: e_ab_pos, w=matrix[src,dst]
        {12, feature_a, feature_b, NA, NB, 0},  // m2: e_ab_neg
        {13, feature_a, feature_a, NA, NA, 0},  // m3: e_aa_pos
        {14, feature_a, feature_a, NA, NA, 0},  // m4: e_aa_neg
        {15, feature_b, feature_a, NB, NA, 2},  // m5: e_ba_pos, w=matrix[dst,src]
        {16, feature_b, feature_a, NB, NA, 0},  // m6: e_ba_neg
        {17, feature_b, feature_b, NB, NB, 0},  // m7: e_bb_pos
        {18, feature_b, feature_b, NB, NB, 0},  // m8: e_bb_neg
    };

    for (int k = 0; k < 8; ++k) {
        const int* edges = (const int*)d_in[cfg[k].ei];
        const int  E     = in_sizes[cfg[k].ei] / 2;
        const int  tilesE   = (cfg[k].Nemb + 15) / 16;
        const int  segPad   = ((cfg[k].Nseg + 15) / 16) * 16;

        // new_emb = femb @ W_agg[k].T + b_agg[k]
        transform_kernel<<<(tilesE * 4 + 7) / 8, 256, 0, stream>>>(
            cfg[k].femb, W_agg + k * 64 * 64, b_agg + k * 64,
            tmp_emb, cfg[k].Nemb, tilesE);

        // zero accumulators (num lives in mbuf[k]; padded rows stay 0)
        long long nz = (long long)segPad * 64;
        zero_kernel<<<(int)((nz + 255) / 256), 256, 0, stream>>>(mbuf[k], nz);
        zero_kernel<<<(segPad + 255) / 256, 256, 0, stream>>>(den, segPad);

        // per-edge attention + segment scatter-add
        edge_agg_kernel<<<(E + 7) / 8, 256, 0, stream>>>(
            edges, E, cfg[k].fsrc, tmp_emb, a_agg + k * 128,
            matrix, cfg[k].wmode, NB, mbuf[k], den);

        // m = num / max(den, den==0 -> 1)
        long long nf = (long long)cfg[k].Nseg * 64;
        finalize_kernel<<<(int)((nf + 255) / 256), 256, 0, stream>>>(
            mbuf[k], den, cfg[k].Nseg);
    }

    // new_a / new_b via fused WMMA MLP
    update_kernel<<<tilesA, 256, 0, stream>>>(
        feature_a, mbuf[0], mbuf[1], mbuf[2], mbuf[3],
        W1, b1, alpha, W2, b2, (float*)d_out, NA);
    update_kernel<<<tilesB, 256, 0, stream>>>(
        feature_b, mbuf[4], mbuf[5], mbuf[6], mbuf[7],
        W1, b1, alpha, W2, b2, (float*)d_out + (size_t)NA * 64, NB);
}